// ModuleSoftsplat_283467842146
// MI455X (gfx1250) — compile-verified
//
#include <hip/hip_runtime.h>

// Problem shape (fixed by the reference): N=8, C=64, H=288, W=512
#define N_   8
#define C_   64
#define H_   288
#define W_   512
#define HW_   (H_ * W_)        // 147456
#define CHW_  (C_ * HW_)       // 9437184
#define NCHW_ (N_ * CHW_)      // 75497472  (301,989,888 bytes)
#define NHW_  (N_ * HW_)       // 1179648   (norm plane elements)

typedef float    f32x4 __attribute__((ext_vector_type(4)));
typedef unsigned u32x4 __attribute__((ext_vector_type(4)));
typedef int      i32x4 __attribute__((ext_vector_type(4)));
typedef int      i32x8 __attribute__((ext_vector_type(8)));

// ---------------------------------------------------------------------------
// TDM: issue a 2D TENSOR_LOAD_TO_LDS. Tile = tile0 (contiguous elements,
// 4B each) x tile1 rows at row stride `stride0` elements.
// D# layout per CDNA5 ISA ch.8:
//   group0: [1:0]=count=1 | lds_addr[63:32] | global_addr[120:64] | type[127:126]=2
//   group1: data_size[17:16]=2 (4B) | tensor_dim0[79:48] | tensor_dim1[111:80]
//           | tile_dim0[127:112] | tile_dim1[143:128] | tensor_dim0_stride[207:160]
//   group2/3: zero (tensor is 2-D; dims 2..4 unused)
// This toolchain exposes the 6-arg builtin (extra int32x8 operand + cpol).
// ---------------------------------------------------------------------------
__device__ __forceinline__ void tdm_load_2d(unsigned lds_off, const float* gptr,
                                            unsigned tile0, unsigned tile1,
                                            unsigned td0, unsigned td1,
                                            unsigned stride0) {
    unsigned long long ga = (unsigned long long)(uintptr_t)gptr;
    u32x4 g0;
    g0.x = 1u;                                                    // count=1, user D#
    g0.y = lds_off;                                               // LDS byte address
    g0.z = (unsigned)ga;                                          // global addr [31:0]
    g0.w = (unsigned)((ga >> 32) & 0x1FFFFFFu) | 0x80000000u;     // addr[56:32] | type=2
    i32x8 g1;
    g1[0] = (int)0x20000u;                                        // data_size = 4 bytes
    g1[1] = (int)((td0 & 0xFFFFu) << 16);                         // tensor_dim0[15:0]
    g1[2] = (int)((td0 >> 16) | ((td1 & 0xFFFFu) << 16));         // td0[31:16] | td1[15:0]
    g1[3] = (int)((td1 >> 16) | (tile0 << 16));                   // td1[31:16] | tile_dim0
    g1[4] = (int)(tile1 & 0xFFFFu);                               // tile_dim1 (tile_dim2=0)
    g1[5] = (int)stride0;                                         // dim0_stride[31:0]
    g1[6] = 0;                                                    // dim0_stride[47:32]=0
    g1[7] = 0;                                                    // dim1_stride=0 (2-D)
    i32x4 gz4 = {0, 0, 0, 0};
    i32x8 gz8 = {0, 0, 0, 0, 0, 0, 0, 0};
    __builtin_amdgcn_tensor_load_to_lds(g0, g1, gz4, gz4, gz8, 0);
}

// ---------------------------------------------------------------------------
// Kernel 1: zero accumulators (output numerators + norm plane in ws).
// Regular-temporal stores on purpose: keep freshly-zeroed lines resident in L2
// so the scatter pass's atomics hit in-cache.
// ---------------------------------------------------------------------------
__global__ __launch_bounds__(256) void zero_f4(f32x4* __restrict__ out4,
                                               f32x4* __restrict__ norm4) {
    unsigned g = blockIdx.x * 256u + threadIdx.x;
    const unsigned n_out4  = NCHW_ / 4;   // 18,874,368
    const unsigned n_norm4 = NHW_ / 4;    //    294,912
    const f32x4 z = {0.f, 0.f, 0.f, 0.f};
    if (g < n_out4) {
        out4[g] = z;
    } else if (g < n_out4 + n_norm4) {
        norm4[g - n_out4] = z;
    }
}

// ---------------------------------------------------------------------------
// Kernel 2: forward splat. grid = (HW/256, N); one thread per source pixel.
// Per wave: 32 contiguous pixels. Input channels are DMA'd by the TDM as two
// 32x32 tiles (channels [0,32) and [32,64)) into this wave's private 8 KB of
// LDS; the VMEM port is left entirely to the no-return global_atomic_add_f32
// stream. Half 0 is processed (s_wait_tensorcnt 1) while half 1 is in flight.
// Corner-validity guards are applied once per 8 atomics (loop-invariant).
// ---------------------------------------------------------------------------
__global__ __launch_bounds__(256) void splat(const float* __restrict__ in,
                                             const float* __restrict__ flow,
                                             const float* __restrict__ metric,
                                             float* __restrict__ out,
                                             float* __restrict__ norm) {
    __shared__ float smem[8 * 2048];                 // 8 waves x (64ch x 32px) = 64 KB

    const unsigned tid = threadIdx.x;
    const unsigned wv  = __builtin_amdgcn_readfirstlane(tid >> 5);  // wave id (uniform)
    const unsigned ln  = tid & 31u;
    const unsigned n   = blockIdx.y;
    const unsigned pixw = blockIdx.x * 256u + wv * 32u;             // wave base pixel
    const unsigned pix  = pixw + ln;                                // lane pixel
    const unsigned y   = pix >> 9;                                  // W = 512
    const unsigned x   = pix & 511u;

    // Kick off both input-tile DMAs first: TDM runs while we do the VALU work.
    const unsigned ldsw = (unsigned)(uintptr_t)&smem[0] + wv * 8192u;
    const float*   gsrc = in + (size_t)n * CHW_ + pixw;
    tdm_load_2d(ldsw,         gsrc,                    32u, 32u, HW_, C_, HW_);
    tdm_load_2d(ldsw + 4096u, gsrc + (size_t)32 * HW_, 32u, 32u, HW_, C_, HW_);

    const float* fl = flow + (size_t)n * 2u * HW_ + pix;
    float fx = (float)x + fl[0];       // flow[:,0]
    float fy = (float)y + fl[HW_];     // flow[:,1]
    bool finite = __builtin_isfinite(fx) && __builtin_isfinite(fy);

    float x0f = floorf(fx), y0f = floorf(fy);
    int ix0 = (int)x0f, iy0 = (int)y0f;
    int ix1 = ix0 + 1,  iy1 = iy0 + 1;
    float wx1 = fx - x0f, wx0 = x0f + 1.0f - fx;
    float wy1 = fy - y0f, wy0 = y0f + 1.0f - fy;

    float w = __expf(metric[(size_t)n * HW_ + pix]);  // softmax weight

    const float w00 = wx0 * wy0 * w;
    const float w10 = wx1 * wy0 * w;
    const float w01 = wx0 * wy1 * w;
    const float w11 = wx1 * wy1 * w;

    const bool vx0 = (ix0 >= 0) & (ix0 < W_);
    const bool vx1 = (ix1 >= 0) & (ix1 < W_);
    const bool vy0 = (iy0 >= 0) & (iy0 < H_);
    const bool vy1 = (iy1 >= 0) & (iy1 < H_);
    const bool v00 = finite & vx0 & vy0;
    const bool v10 = finite & vx1 & vy0;
    const bool v01 = finite & vx0 & vy1;
    const bool v11 = finite & vx1 & vy1;

    const int b00 = iy0 * W_ + ix0;
    const int b10 = iy0 * W_ + ix1;
    const int b01 = iy1 * W_ + ix0;
    const int b11 = iy1 * W_ + ix1;

    // Denominator plane, once per pixel.
    float* nb = norm + (size_t)n * HW_;
    if (v00) atomicAdd(nb + b00, w00);
    if (v10) atomicAdd(nb + b10, w10);
    if (v01) atomicAdd(nb + b01, w01);
    if (v11) atomicAdd(nb + b11, w11);

    float*       obase = out + (size_t)n * CHW_;
    const float* sw    = &smem[wv * 2048u + ln];      // this lane's LDS column

    auto process_half = [&](int h) {
        const float* sh = sw + h * 1024;
        for (int c0 = 0; c0 < 32; c0 += 8) {
            float v[8];
#pragma unroll
            for (int j = 0; j < 8; ++j) v[j] = sh[(c0 + j) * 32];   // ds_load_b32, no conflicts

            float* oc = obase + (size_t)(h * 32 + c0) * HW_;
            if (v00) {
#pragma unroll
                for (int j = 0; j < 8; ++j) atomicAdd(oc + (size_t)j * HW_ + b00, v[j] * w00);
            }
            if (v10) {
#pragma unroll
                for (int j = 0; j < 8; ++j) atomicAdd(oc + (size_t)j * HW_ + b10, v[j] * w10);
            }
            if (v01) {
#pragma unroll
                for (int j = 0; j < 8; ++j) atomicAdd(oc + (size_t)j * HW_ + b01, v[j] * w01);
            }
            if (v11) {
#pragma unroll
                for (int j = 0; j < 8; ++j) atomicAdd(oc + (size_t)j * HW_ + b11, v[j] * w11);
            }
        }
    };

    // Half 0: wait until only the second TDM op is outstanding (in-order cnt).
    __builtin_amdgcn_s_wait_tensorcnt(1);
    asm volatile("" ::: "memory");                   // keep LDS reads below the wait
    process_half(0);

    // Half 1: wait for all tensor ops.
    __builtin_amdgcn_s_wait_tensorcnt(0);
    asm volatile("" ::: "memory");
    process_half(1);
}

// ---------------------------------------------------------------------------
// Kernel 3: normalize out[n,c,y,x] /= (norm[n,y,x] == 0 ? 1 : norm[n,y,x]).
// grid = (HW/(4*256), C, N). float4 vectorized; norm plane (4.5 MB, reused
// 64x) stays regular-temporal; out is read-once/write-once -> NT both ways.
// ---------------------------------------------------------------------------
__global__ __launch_bounds__(256) void normalize_k(float* __restrict__ out,
                                                   const float* __restrict__ norm) {
    const unsigned pix = (blockIdx.x * 256u + threadIdx.x) * 4u;  // over H*W
    const unsigned c   = blockIdx.y;
    const unsigned n   = blockIdx.z;

    const f32x4 nv = *reinterpret_cast<const f32x4*>(norm + (size_t)n * HW_ + pix);
    f32x4* po = reinterpret_cast<f32x4*>(out + (size_t)(n * C_ + c) * HW_ + pix);

    f32x4 o = __builtin_nontemporal_load(po);
    o.x = o.x / ((nv.x == 0.f) ? 1.f : nv.x);
    o.y = o.y / ((nv.y == 0.f) ? 1.f : nv.y);
    o.z = o.z / ((nv.z == 0.f) ? 1.f : nv.z);
    o.w = o.w / ((nv.w == 0.f) ? 1.f : nv.w);
    __builtin_nontemporal_store(o, po);
}

// ---------------------------------------------------------------------------
extern "C" void kernel_launch(void* const* d_in, const int* in_sizes, int n_in,
                              void* d_out, int out_size, void* d_ws, size_t ws_size,
                              hipStream_t stream) {
    (void)in_sizes; (void)n_in; (void)out_size; (void)ws_size;

    const float* tenIn     = (const float*)d_in[0];   // [8,64,288,512]
    const float* tenFlow   = (const float*)d_in[1];   // [8,2,288,512]
    const float* tenMetric = (const float*)d_in[2];   // [8,1,288,512]
    float*       out       = (float*)d_out;           // [8,64,288,512]
    float*       normPlane = (float*)d_ws;            // [8,288,512] = 4.5 MB of ws

    // 1) zero accumulators
    {
        const unsigned total4 = NCHW_ / 4 + NHW_ / 4;   // 19,169,280
        const unsigned blocks = (total4 + 255u) / 256u; // 74,880
        zero_f4<<<blocks, 256, 0, stream>>>((f32x4*)out, (f32x4*)normPlane);
    }
    // 2) scatter (TDM-staged inputs)
    {
        dim3 grid(HW_ / 256, N_);                       // (576, 8)
        splat<<<grid, 256, 0, stream>>>(tenIn, tenFlow, tenMetric, out, normPlane);
    }
    // 3) normalize
    {
        dim3 grid(HW_ / (4 * 256), C_, N_);             // (144, 64, 8)
        normalize_k<<<grid, 256, 0, stream>>>(out, normPlane);
    }
}